// MultiHeadsAttModel_80444737454734
// MI455X (gfx1250) — compile-verified
//
#include <hip/hip_runtime.h>
#include <math.h>

typedef float v2f __attribute__((ext_vector_type(2)));
typedef float v8f __attribute__((ext_vector_type(8)));

#define MAXN 0.996f     /* (1-EPS)/sqrt(c), c=1, EPS=4e-3 */
#define MINN 1e-15f

#define Bdim 4
#define Ndim 2000
#define Kn   5
#define Ddim 32
#define H1   160
#define HEAD 5
#define AG_PER_BLK 8          /* agents per wave */
#define ROWS (AG_PER_BLK*(1+Kn))   /* 48 rows: 8 agents + 40 neighbors */

__device__ __forceinline__ float wred(float v) {
#pragma unroll
  for (int o = 16; o; o >>= 1) v += __shfl_xor(v, o, 32);
  return v;
}

__device__ __forceinline__ float artanhf_(float x) {
  x = fminf(fmaxf(x, -1.0f + 1e-7f), 1.0f - 1e-7f);
  return 0.5f * (log1pf(x) - log1pf(-x));
}

// ---------------------------------------------------------------------------
// Kernel 1: recover neighbor index from one-hot rows. One wave per row.
// Reads 320 MB coalesced (the dominant cost of the whole op, ~14us @ 23.3TB/s)
// ---------------------------------------------------------------------------
__global__ void onehot_argmax_kernel(const float* __restrict__ nei,
                                     int* __restrict__ idx, int nrows) {
  int wid  = (blockIdx.x * blockDim.x + threadIdx.x) >> 5;
  int lane = threadIdx.x & 31;
  if (wid >= nrows) return;
  const float4* row = (const float4*)(nei + (size_t)wid * Ndim);
  int found = -1;
#pragma unroll 4
  for (int j = lane; j < Ndim / 4; j += 32) {
    float4 v = row[j];
    if (v.x > 0.5f) found = 4 * j + 0;
    if (v.y > 0.5f) found = 4 * j + 1;
    if (v.z > 0.5f) found = 4 * j + 2;
    if (v.w > 0.5f) found = 4 * j + 3;
  }
#pragma unroll
  for (int o = 16; o; o >>= 1) found = max(found, __shfl_xor(found, o, 32));
  if (lane == 0) idx[wid] = max(found, 0);
}

// ---------------------------------------------------------------------------
// Kernel 2: fused hyperbolic attention. One wave (32 threads) per 8 agents.
// ---------------------------------------------------------------------------
__global__ void __launch_bounds__(32)
hyp_att_kernel(const float* __restrict__ feats_in,  // [B,N,32] points on ball
               const int* __restrict__ nbr_idx,     // [B*N*K]
               const float* __restrict__ W1,        // [160,32]
               const float* __restrict__ b1,        // [160]
               const float* __restrict__ W2,        // [32,32]
               const float* __restrict__ b2,        // [32]
               float* __restrict__ out,             // [B*N*32]
               float* __restrict__ att_out) {       // [B*N*5*5]
  __shared__ float Xs[ROWS][Ddim];     // expmap'd input rows (A matrix stage1)
  __shared__ float Ys[ROWS][H1];       // mx, later q/kv tangent rows
  __shared__ float xns[ROWS];          // ||x|| per input row
  __shared__ float HB1[H1];
  __shared__ float HB2[Ddim];
  __shared__ float X2s[16][Ddim];      // A matrix stage2 (8 used + 8 zero)
  __shared__ float MX2[16][Ddim];      // stage2 WMMA result
  __shared__ float xn2[16];
  __shared__ float LOGI[HEAD * Kn];
  __shared__ float ATTW[HEAD * Kn];
  __shared__ float scal[2];            // ||hb1||^2, ||hb2||^2

  const int lane = threadIdx.x;
  const int g0   = blockIdx.x * AG_PER_BLK;   // flat agent base over B*N

  // ---- hyperbolic biases: hb = proj(expmap0(b)) ----
  {
    float v[5], s = 0.f;
#pragma unroll
    for (int t = 0; t < 5; ++t) { v[t] = b1[lane + 32 * t]; s += v[t] * v[t]; }
    float n2 = wred(s);
    float rawn = sqrtf(n2), n = fmaxf(rawn, MINN);
    float es = tanhf(n) / n, en = es * rawn;
    if (en > MAXN) { es *= MAXN / en; en = MAXN; }
#pragma unroll
    for (int t = 0; t < 5; ++t) HB1[lane + 32 * t] = es * v[t];
    if (lane == 0) scal[0] = en * en;

    float w = b2[lane];
    float m2 = wred(w * w);
    float rb = sqrtf(m2), nb = fmaxf(rb, MINN);
    float eb = tanhf(nb) / nb, enb = eb * rb;
    if (enb > MAXN) { eb *= MAXN / enb; enb = MAXN; }
    HB2[lane] = eb * w;
    if (lane == 0) scal[1] = enb * enb;
  }
  __syncthreads();

  // ---- Phase A: build 48 input rows x = proj(expmap0(logmap0(p))) ----
  for (int r = 0; r < ROWS; ++r) {
    int srcg;
    if (r < AG_PER_BLK) {
      srcg = g0 + r;
    } else {
      int m = (r - AG_PER_BLK) / Kn, k = (r - AG_PER_BLK) % Kn;
      int g = g0 + m;
      int b = g / Ndim;
      srcg = b * Ndim + nbr_idx[g * Kn + k];
    }
    float f   = feats_in[(size_t)srcg * Ddim + lane];
    float n2  = wred(f * f);
    float rawn = sqrtf(n2), nn = fmaxf(rawn, MINN);
    float at  = artanhf_(nn);                 // logmap0 numerator
    float ls  = at / nn;                      // logmap0 scale
    float nu  = fmaxf(ls * rawn, MINN);       // ||u||
    float scale = ls * (tanhf(nu) / nu);      // expmap0(logmap0) combined
    float xnorm = scale * rawn;
    if (xnorm > MAXN) { scale *= MAXN / xnorm; xnorm = MAXN; }
    Xs[r][lane] = scale * f;
    if (lane == 0) xns[r] = fmaxf(xnorm, MINN);
  }
  __syncthreads();

  // ---- Phase B: mx = X @ W1^T via V_WMMA_F32_16X16X4_F32 ----
  {
    const int khalf = (lane < 16) ? 0 : 2;
    const int l16   = lane & 15;
    for (int nt = 0; nt < H1 / 16; ++nt) {
      v2f btile[8];
      int N = nt * 16 + l16;
#pragma unroll
      for (int ks = 0; ks < 8; ++ks) {        // B[k][n] = W1[n][k]
        btile[ks].x = W1[N * Ddim + ks * 4 + khalf];
        btile[ks].y = W1[N * Ddim + ks * 4 + khalf + 1];
      }
#pragma unroll
      for (int rt = 0; rt < ROWS / 16; ++rt) {
        v8f cacc{};
        int M = rt * 16 + l16;
#pragma unroll
        for (int ks = 0; ks < 8; ++ks) {
          v2f a;
          a.x = Xs[M][ks * 4 + khalf];
          a.y = Xs[M][ks * 4 + khalf + 1];
          cacc = __builtin_amdgcn_wmma_f32_16x16x4_f32(
              false, a, false, btile[ks], (short)0, cacc, false, false);
        }
        int rbase = rt * 16 + ((lane < 16) ? 0 : 8);
#pragma unroll
        for (int j = 0; j < 8; ++j) Ys[rbase + j][nt * 16 + l16] = cacc[j];
      }
    }
  }
  __syncthreads();

  // ---- Phase C: per-row hyperbolic pipeline on 160-dim rows, then logmap0 ----
  const float y2g = scal[0];
  for (int r = 0; r < ROWS; ++r) {
    float v[5], s = 0.f, hd = 0.f;
#pragma unroll
    for (int t = 0; t < 5; ++t) {
      v[t] = Ys[r][lane + 32 * t];
      s  += v[t] * v[t];
      hd += v[t] * HB1[lane + 32 * t];
    }
    float mxn2 = wred(s);
    float dotraw = wred(hd);
    bool  zero = (mxn2 == 0.f);
    float mraw = sqrtf(mxn2), mxn = fmaxf(mraw, MINN);
    float xr = xns[r];
    float t1 = tanhf(mxn / xr * artanhf_(xr));      // mobius_matvec
    float sc1 = t1 / mxn, rn = t1;
    if (rn > MAXN) { sc1 *= MAXN / rn; rn = MAXN; } // proj
    if (zero) { sc1 = 0.f; rn = 0.f; }
    // mobius_add(res, hb1)
    float xy = sc1 * dotraw, x2 = rn * rn;
    float A = 1.f + 2.f * xy + y2g, Bc = 1.f - x2;
    float den = fmaxf(1.f + 2.f * xy + x2 * y2g, MINN);
    float alpha = A * sc1 / den, beta = Bc / den;
    float rn2 = sqrtf(fmaxf((A * A * x2 + 2.f * A * Bc * xy + Bc * Bc * y2g) / (den * den), 0.f));
    if (rn2 > MAXN) { float p = MAXN / rn2; alpha *= p; beta *= p; rn2 = MAXN; }
    // relu(logmap0(.))
    float nl = fmaxf(rn2, MINN);
    float lsc = artanhf_(nl) / nl;
    float xt[5], s2 = 0.f;
#pragma unroll
    for (int t = 0; t < 5; ++t) {
      float r2 = alpha * v[t] + beta * HB1[lane + 32 * t];
      float x = fmaxf(lsc * r2, 0.f);
      xt[t] = x; s2 += x * x;
    }
    // proj(expmap0(.)) then logmap0 for q/kv
    float nxt2 = wred(s2);
    float rxt = sqrtf(nxt2), nxt = fmaxf(rxt, MINN);
    float es = tanhf(nxt) / nxt, en = es * rxt;
    if (en > MAXN) { es *= MAXN / en; en = MAXN; }
    float n3 = fmaxf(en, MINN);
    float lg = artanhf_(n3) / n3;
#pragma unroll
    for (int t = 0; t < 5; ++t) Ys[r][lane + 32 * t] = lg * es * xt[t];
  }
  __syncthreads();

  // ---- Phase D: 5-head attention over K=5 neighbors per agent ----
  for (int m = 0; m < AG_PER_BLK; ++m) {
    const int g = g0 + m;
    const int h = lane / Kn, k = lane % Kn;
    const bool act = (lane < HEAD * Kn);
    float dot = 0.f;
    if (act) {
#pragma unroll
      for (int d = 0; d < Ddim; ++d)
        dot += Ys[m][d * HEAD + h] * Ys[AG_PER_BLK + m * Kn + k][d * HEAD + h];
      LOGI[lane] = dot;
    }
    __syncthreads();
    if (act) {
      float mx = -1e30f;
#pragma unroll
      for (int kk = 0; kk < Kn; ++kk) mx = fmaxf(mx, LOGI[h * Kn + kk]);
      float se = 0.f;
#pragma unroll
      for (int kk = 0; kk < Kn; ++kk) se += expf(LOGI[h * Kn + kk] - mx);
      float w = expf(dot - mx) / se;
      ATTW[lane] = w;
      att_out[(g * HEAD + h) * Kn + k] = w;
    }
    __syncthreads();
    // out[d] = mean_h sum_k att*kv
    float o = 0.f;
#pragma unroll
    for (int hh = 0; hh < HEAD; ++hh) {
      float acc = 0.f;
#pragma unroll
      for (int kk = 0; kk < Kn; ++kk)
        acc += ATTW[hh * Kn + kk] * Ys[AG_PER_BLK + m * Kn + kk][lane * HEAD + hh];
      o += acc;
    }
    o *= (1.0f / HEAD);
    // proj(expmap0(out))
    float n2 = wred(o * o);
    float rawn = sqrtf(n2), n = fmaxf(rawn, MINN);
    float es = tanhf(n) / n, en = es * rawn;
    if (en > MAXN) { es *= MAXN / en; en = MAXN; }
    X2s[m][lane] = es * o;
    if (lane == 0) xn2[m] = fmaxf(en, MINN);
    __syncthreads();
  }
  for (int r = AG_PER_BLK; r < 16; ++r) {
    X2s[r][lane] = 0.f;
    if (lane == 0) xn2[r] = 1.f;
  }
  __syncthreads();

  // ---- Phase E: final hnn_layer with W2 via WMMA (16x32 @ 32x32) ----
  {
    const int khalf = (lane < 16) ? 0 : 2;
    const int l16 = lane & 15;
#pragma unroll
    for (int nt = 0; nt < 2; ++nt) {
      int N = nt * 16 + l16;
      v8f cacc{};
#pragma unroll
      for (int ks = 0; ks < 8; ++ks) {
        v2f a, b;
        a.x = X2s[l16][ks * 4 + khalf];
        a.y = X2s[l16][ks * 4 + khalf + 1];
        b.x = W2[N * Ddim + ks * 4 + khalf];
        b.y = W2[N * Ddim + ks * 4 + khalf + 1];
        cacc = __builtin_amdgcn_wmma_f32_16x16x4_f32(
            false, a, false, b, (short)0, cacc, false, false);
      }
      int rbase = (lane < 16) ? 0 : 8;
#pragma unroll
      for (int j = 0; j < 8; ++j) MX2[rbase + j][N] = cacc[j];
    }
  }
  __syncthreads();

  // ---- Phase F: post-process 32-dim rows, write output ----
  const float y2b = scal[1];
  for (int r = 0; r < AG_PER_BLK; ++r) {
    float v = MX2[r][lane];
    float mxn2 = wred(v * v);
    float dotraw = wred(v * HB2[lane]);
    bool zero = (mxn2 == 0.f);
    float mraw = sqrtf(mxn2), mxn = fmaxf(mraw, MINN);
    float xr = xn2[r];
    float t1 = tanhf(mxn / xr * artanhf_(xr));
    float sc1 = t1 / mxn, rn = t1;
    if (rn > MAXN) { sc1 *= MAXN / rn; rn = MAXN; }
    if (zero) { sc1 = 0.f; rn = 0.f; }
    float xy = sc1 * dotraw, x2 = rn * rn;
    float A = 1.f + 2.f * xy + y2b, Bc = 1.f - x2;
    float den = fmaxf(1.f + 2.f * xy + x2 * y2b, MINN);
    float alpha = A * sc1 / den, beta = Bc / den;
    float rn2 = sqrtf(fmaxf((A * A * x2 + 2.f * A * Bc * xy + Bc * Bc * y2b) / (den * den), 0.f));
    if (rn2 > MAXN) { float p = MAXN / rn2; alpha *= p; beta *= p; rn2 = MAXN; }
    float nl = fmaxf(rn2, MINN);
    float lsc = artanhf_(nl) / nl;
    float xt = fmaxf(lsc * (alpha * v + beta * HB2[lane]), 0.f);
    float nxt2 = wred(xt * xt);
    float rxt = sqrtf(nxt2), nxt = fmaxf(rxt, MINN);
    float es = tanhf(nxt) / nxt, en = es * rxt;
    if (en > MAXN) { es *= MAXN / en; }
    out[(size_t)(g0 + r) * Ddim + lane] = es * xt;
  }
}

// ---------------------------------------------------------------------------
extern "C" void kernel_launch(void* const* d_in, const int* in_sizes, int n_in,
                              void* d_out, int out_size, void* d_ws, size_t ws_size,
                              hipStream_t stream) {
  const float* in_feats = (const float*)d_in[0];
  const float* in_nei   = (const float*)d_in[1];
  const float* W1       = (const float*)d_in[2];
  const float* b1       = (const float*)d_in[3];
  const float* W2       = (const float*)d_in[4];
  const float* b2       = (const float*)d_in[5];
  float* out     = (float*)d_out;
  float* att_out = out + (size_t)Bdim * Ndim * Ddim;   // 256000 offset
  int*   idx     = (int*)d_ws;                          // 40000 ints

  const int nrows = Bdim * Ndim * Kn;                   // 40000 one-hot rows
  // 4 waves per 128-thread block, 1 wave per row
  onehot_argmax_kernel<<<(nrows * 32 + 127) / 128, 128, 0, stream>>>(in_nei, idx, nrows);

  const int nblocks = (Bdim * Ndim) / AG_PER_BLK;       // 1000
  hyp_att_kernel<<<nblocks, 32, 0, stream>>>(in_feats, idx, W1, b1, W2, b2, out, att_out);
}